// QuantumVocabularyManager_3977139716533
// MI455X (gfx1250) — compile-verified
//
#include <hip/hip_runtime.h>
#include <math.h>

// Cosine-similarity top-k (QuantumVocabularyManager) for gfx1250 / MI455X.
//
// Roofline: vocab = 128000 x 1024 f32 = 524 MB streamed once -> ~22.5 us at
// 23.3 TB/s. Compute is ~0.5 GFLOP -> pure bandwidth problem. We therefore
// fuse ||row||^2 and row.psi into a single pass over HBM, keep everything
// f32, and run the dot products through V_WMMA_F32_16X16X4_F32 (the native
// f32 WMMA on CDNA5). WMMA issue cost (~2M instructions device-wide) is far
// below the memory floor, so using only column 0 of B is free.

#define EPS       1e-9f
#define D_DIM     1024     // embed dim (reference)
#define ROWS_PER_WAVE  16  // WMMA M
#define WAVES_PER_BLK   4
#define ROWS_PER_BLK   64  // 16 rows/wave * 4 waves
#define TOPK            5

typedef __attribute__((ext_vector_type(2))) float v2f;
typedef __attribute__((ext_vector_type(8))) float v8f;

// ---------------------------------------------------------------- kernel 0
// psi_out = psi / (||psi|| + eps)
__global__ __launch_bounds__(256)
void psi_normalize_kernel(const float* __restrict__ psi,
                          float* __restrict__ psi_out, int d) {
    __shared__ float red[256];
    const int t = threadIdx.x;
    float s = 0.f;
    for (int i = t; i < d; i += 256) { float x = psi[i]; s += x * x; }
    red[t] = s;
    __syncthreads();
    for (int off = 128; off > 0; off >>= 1) {
        if (t < off) red[t] += red[t + off];
        __syncthreads();
    }
    const float inv = 1.0f / (sqrtf(red[0]) + EPS);
    for (int i = t; i < d; i += 256) psi_out[i] = psi[i] * inv;
}

// ---------------------------------------------------------------- kernel 1
// Each wave: 16 vocab rows, full K. Single HBM pass per element:
//  - A chunk (16x4) loaded per-lane directly in WMMA A layout
//    (lane 0-15 = rows, lane half picks K{0,1} vs K{2,3}).
//  - B chunk (4x16) = psi[c..c+3] in column 0, zeros elsewhere
//    (B layout assumed to mirror A: VGPR0 = K0|K2, VGPR1 = K1|K3).
//  - row norm fused: lane L + lane L+16 jointly cover row L%16 exactly once,
//    so per-lane sum-of-squares + shfl_xor(16) gives ||row||^2 for free.
__global__ __launch_bounds__(128)
void cosine_topk_block_kernel(const float* __restrict__ vocab,
                              const float* __restrict__ psi_n,
                              float* __restrict__ block_out /*nblk*TOPK*2*/) {
    __shared__ float spsi[D_DIM];
    __shared__ float ssim[ROWS_PER_BLK];

    const int t    = threadIdx.x;
    const int lane = t & 31;
    const int w    = t >> 5;
    const int m    = lane & 15;   // row within the wave tile
    const int half = lane >> 4;   // 0 -> K{0,1}, 1 -> K{2,3}

    // Stage normalized psi in LDS (4 KB), read per WMMA step as ds_load_b64.
    {
        float4*       sp4 = (float4*)spsi;
        const float4* pn4 = (const float4*)psi_n;
        #pragma unroll
        for (int i = 0; i < D_DIM / 4 / 128; ++i)
            sp4[t + i * 128] = pn4[t + i * 128];
    }
    __syncthreads();

    const int rowBase = blockIdx.x * ROWS_PER_BLK + w * ROWS_PER_WAVE;
    const float* aPtr = vocab + (size_t)(rowBase + m) * D_DIM + 2 * half;

    v8f  acc  = {};     // 16x16 f32 accumulator (column 0 is the real dot)
    float nacc = 0.f;   // per-lane partial sum of squares

    #pragma unroll 8
    for (int c = 0; c < D_DIM; c += 4) {
        const v2f a = *(const v2f*)(aPtr + c);                 // global b64
        const v2f p = *(const v2f*)(spsi + c + 2 * half);      // ds b64
        v2f b;
        b.x = (m == 0) ? p.x : 0.0f;   // only column N==0 carries psi
        b.y = (m == 0) ? p.y : 0.0f;
        nacc += a.x * a.x + a.y * a.y;
        // D = A(16x4) * B(4x16) + C  -- v_wmma_f32_16x16x4_f32
        acc = __builtin_amdgcn_wmma_f32_16x16x4_f32(
                  false, a, false, b, (short)0, acc, false, false);
    }

    // Complete ||row||^2: partner lane holds the other half of the columns.
    nacc += __shfl_xor(nacc, 16, 32);

    // Extract D column 0: lane 0 VGPR r = row r, lane 16 VGPR r = row r+8.
    float dot = 0.f;
    #pragma unroll
    for (int r = 0; r < 8; ++r) {
        const float lo = __shfl(acc[r], 0, 32);
        const float hi = __shfl(acc[r], 16, 32);
        if (lane == r)     dot = lo;
        if (lane == r + 8) dot = hi;
    }

    if (lane < 16)
        ssim[w * ROWS_PER_WAVE + lane] = dot / (sqrtf(nacc) + EPS);
    __syncthreads();

    // Block-local top-5 over 64 sims, done by wave 0 (argmax + mark -inf).
    if (w == 0) {
        for (int k = 0; k < TOPK; ++k) {
            float v0 = ssim[lane];
            float v1 = ssim[lane + 32];
            float v; int id;
            if (v1 > v0) { v = v1; id = lane + 32; }
            else         { v = v0; id = lane;      }
            #pragma unroll
            for (int off = 16; off > 0; off >>= 1) {
                const float ov  = __shfl_xor(v,  off, 32);
                const int   oid = __shfl_xor(id, off, 32);
                if (ov > v || (ov == v && oid < id)) { v = ov; id = oid; }
            }
            if (lane == 0) {
                const size_t base = (size_t)blockIdx.x * TOPK * 2 + k * 2;
                block_out[base] = v;
                ((int*)block_out)[base + 1] = blockIdx.x * ROWS_PER_BLK + id;
                ssim[id] = -INFINITY;   // per-wave LDS ops are in-order
            }
        }
    }
}

// ---------------------------------------------------------------- kernel 2
// Reduce nblk*TOPK (score, idx) candidates to the global top-5.
__global__ __launch_bounds__(256)
void final_topk_kernel(float* __restrict__ cand, int nc,
                       float* __restrict__ out) {
    __shared__ float sv[256];
    __shared__ int   si[256];
    const int t = threadIdx.x;
    for (int k = 0; k < TOPK; ++k) {
        float best = -INFINITY; int bi = 0x7fffffff;
        for (int i = t; i < nc; i += 256) {
            const float v = cand[(size_t)i * 2];
            if (v > best) { best = v; bi = i; }
        }
        sv[t] = best; si[t] = bi;
        __syncthreads();
        for (int s = 128; s > 0; s >>= 1) {
            if (t < s) {
                if (sv[t + s] > sv[t] ||
                    (sv[t + s] == sv[t] && si[t + s] < si[t])) {
                    sv[t] = sv[t + s]; si[t] = si[t + s];
                }
            }
            __syncthreads();
        }
        if (t == 0) {
            const int ci = si[0];
            out[k] = sv[0];                                    // f32 scores
            ((int*)out)[TOPK + k] = ((const int*)cand)[(size_t)ci * 2 + 1];
            cand[(size_t)ci * 2] = -INFINITY;                  // exclude
        }
        __syncthreads();
    }
}

// ---------------------------------------------------------------- launch
extern "C" void kernel_launch(void* const* d_in, const int* in_sizes, int n_in,
                              void* d_out, int out_size, void* d_ws,
                              size_t ws_size, hipStream_t stream) {
    const float* psi   = (const float*)d_in[0];
    const float* vocab = (const float*)d_in[1];
    const int d = in_sizes[0];          // 1024 (kernel 1 assumes D_DIM)
    const int v = in_sizes[1] / d;      // 128000 (divisible by 64)

    float* ws        = (float*)d_ws;
    float* psi_n     = ws;              // d floats
    float* block_out = ws + d;          // nblk * TOPK * 2 floats

    const int nblk = v / ROWS_PER_BLK;

    psi_normalize_kernel<<<1, 256, 0, stream>>>(psi, psi_n, d);
    cosine_topk_block_kernel<<<nblk, WAVES_PER_BLK * 32, 0, stream>>>(
        vocab, psi_n, block_out);
    final_topk_kernel<<<1, 256, 0, stream>>>(block_out, nblk * TOPK,
                                             (float*)d_out);
}